// GlobalGraphTransformer_58737972740403
// MI455X (gfx1250) — compile-verified
//
#include <hip/hip_runtime.h>
#include <hip/hip_bf16.h>
#include <cstddef>

// ---------------------------------------------------------------------------
// GlobalGraphTransformer on MI455X (gfx1250, wave32, WMMA f32 16x16x4)
// G=64 graphs x n=256 nodes, H=256, heads=8 (dh=32), L=4, F=512, De=16
// ---------------------------------------------------------------------------

typedef __attribute__((ext_vector_type(2))) float v2f;
typedef __attribute__((ext_vector_type(4))) float v4f;
typedef __attribute__((ext_vector_type(8))) float v8f;
typedef __attribute__((ext_vector_type(4))) int v4i;

#if defined(__gfx1250__) && __has_builtin(__builtin_amdgcn_global_load_async_to_lds_b128)
#define HAVE_ASYNC_LDS 1
#else
#define HAVE_ASYNC_LDS 0
#endif

// Copy 16 bytes global -> LDS. Async (ASYNCcnt, bypasses VGPRs) when available.
static __device__ inline void g2l_16(const float* __restrict__ g, float* l) {
#if HAVE_ASYNC_LDS
    __builtin_amdgcn_global_load_async_to_lds_b128(
        (__attribute__((address_space(1))) v4i*)g,
        (__attribute__((address_space(3))) v4i*)l, 0, 0);
#else
    *(v4f*)l = *(const v4f*)g;
#endif
}

static __device__ inline void wait_async_lds() {
#if HAVE_ASYNC_LDS
#if __has_builtin(__builtin_amdgcn_s_wait_asynccnt)
    __builtin_amdgcn_s_wait_asynccnt(0);
#else
    asm volatile("s_wait_asynccnt 0x0" ::: "memory");
#endif
#endif
}

static __device__ inline v8f vzero8() {
    v8f z = {0.f, 0.f, 0.f, 0.f, 0.f, 0.f, 0.f, 0.f};
    return z;
}

// D = A(16x4) x B(4x16) + C, fp32, wave32.
static __device__ inline v8f wmma4(v2f a, v2f b, v8f c) {
    return __builtin_amdgcn_wmma_f32_16x16x4_f32(
        /*neg_a=*/false, a, /*neg_b=*/false, b,
        /*c_mod=*/(short)0, c, /*reuse_a=*/false, /*reuse_b=*/false);
}

// half-wave (16-lane) reductions; matches the WMMA C-fragment column groups
static __device__ inline float hmax16(float v) {
    for (int o = 8; o > 0; o >>= 1) v = fmaxf(v, __shfl_xor(v, o, 16));
    return v;
}
static __device__ inline float hsum16(float v) {
    for (int o = 8; o > 0; o >>= 1) v += __shfl_xor(v, o, 16);
    return v;
}

// ---------------------------------------------------------------------------
// Edge bias scatter: bias[g,h,sl,dl] += edge_attr[e] . edge_proj_w[h] + b[h]
// ---------------------------------------------------------------------------
__global__ __launch_bounds__(256) void edge_bias_kernel(
    const int* __restrict__ edge_index, const float* __restrict__ edge_attr,
    const float* __restrict__ pw, const float* __restrict__ pb,
    const int* __restrict__ batch_vec, const int* __restrict__ ptr,
    float* __restrict__ bias, int E)
{
    int e = blockIdx.x * blockDim.x + threadIdx.x;
    if (e >= E) return;
    int src = edge_index[e];
    int dst = edge_index[E + e];
    int g = batch_vec[src];
    int p = ptr[g];
    int sl = src - p;
    int dl = dst - p;
    float ea[16];
#pragma unroll
    for (int j = 0; j < 16; ++j) ea[j] = edge_attr[(size_t)e * 16 + j];
#pragma unroll
    for (int h = 0; h < 8; ++h) {
        float acc = pb[h];
#pragma unroll
        for (int j = 0; j < 16; ++j) acc += ea[j] * pw[h * 16 + j];
        atomicAdd(&bias[(((size_t)(g * 8 + h)) * 256 + sl) * 256 + dl], acc);
    }
}

// ---------------------------------------------------------------------------
// Row LayerNorm: one 256-thread block per row of [*, 256]
// ---------------------------------------------------------------------------
__global__ __launch_bounds__(256) void ln_kernel(
    const float* __restrict__ x, const float* __restrict__ gamma,
    const float* __restrict__ beta, float* __restrict__ y)
{
    __shared__ float red[256];
    int row = blockIdx.x;
    int t = threadIdx.x;
    float v = x[(size_t)row * 256 + t];
    red[t] = v;
    __syncthreads();
    for (int s = 128; s > 0; s >>= 1) {
        if (t < s) red[t] += red[t + s];
        __syncthreads();
    }
    float mean = red[0] * (1.f / 256.f);
    __syncthreads();
    float d = v - mean;
    red[t] = d * d;
    __syncthreads();
    for (int s = 128; s > 0; s >>= 1) {
        if (t < s) red[t] += red[t + s];
        __syncthreads();
    }
    float var = red[0] * (1.f / 256.f);
    y[(size_t)row * 256 + t] = d * rsqrtf(var + 1e-5f) * gamma[t] + beta[t];
}

// ---------------------------------------------------------------------------
// fp32 WMMA GEMM: C[M,N] = A[M,K] * W[N,K]^T + bias[N]
//   (+ optional residual add, optional ReLU).  M%64==0, N%128==0, K%32==0.
// Block: 256 threads (8 waves), 64x128 tile; each wave 32x32 (2x2 WMMA tiles)
// so every A/B fragment load feeds two v_wmma (load:wmma ratio 1:1).
// LDS row pad = 36 floats: 16B-aligned rows for async B128 and conflict-free
// fragment reads (36*r mod 64 distinct over 16 rows).
// ---------------------------------------------------------------------------
__global__ __launch_bounds__(256) void gemm_wmma(
    const float* __restrict__ A, const float* __restrict__ W,
    const float* __restrict__ bias, const float* __restrict__ resid,
    float* __restrict__ C, int M, int N, int K, int do_relu)
{
    __shared__ __align__(16) float As[64][36];
    __shared__ __align__(16) float Bs[128][36];
    int tid = threadIdx.x;
    int lane = tid & 31;
    int wave = tid >> 5;
    int m0 = blockIdx.x * 64;
    int n0 = blockIdx.y * 128;
    int wm = (wave >> 2) * 32;  // wave row offset in block tile (0/32)
    int wn = (wave & 3) * 32;   // wave col offset in block tile (0..96)

    v8f acc[2][2];
    acc[0][0] = vzero8(); acc[0][1] = vzero8();
    acc[1][0] = vzero8(); acc[1][1] = vzero8();

    int arow = tid >> 2;        // 0..63   A staging row
    int ak = (tid & 3) * 8;     // 0,8,16,24
    int brow = tid >> 1;        // 0..127  B staging row
    int bk = (tid & 1) * 16;    // 0,16
    int am = lane & 15;         // fragment row/col
    int kb = (lane >> 4) * 2;   // K sub-offset per lane half

    for (int k0 = 0; k0 < K; k0 += 32) {
        const float* ap = A + (size_t)(m0 + arow) * K + k0 + ak;
        const float* bp = W + (size_t)(n0 + brow) * K + k0 + bk;
        g2l_16(ap,     &As[arow][ak]);
        g2l_16(ap + 4, &As[arow][ak + 4]);
#pragma unroll
        for (int j = 0; j < 4; ++j) g2l_16(bp + j * 4, &Bs[brow][bk + j * 4]);
        wait_async_lds();
        __syncthreads();
#pragma unroll
        for (int kc = 0; kc < 8; ++kc) {
            v2f a0, a1, b0, b1;
            a0.x = As[wm + am][kc * 4 + kb];      a0.y = As[wm + am][kc * 4 + kb + 1];
            a1.x = As[wm + 16 + am][kc * 4 + kb]; a1.y = As[wm + 16 + am][kc * 4 + kb + 1];
            b0.x = Bs[wn + am][kc * 4 + kb];      b0.y = Bs[wn + am][kc * 4 + kb + 1];
            b1.x = Bs[wn + 16 + am][kc * 4 + kb]; b1.y = Bs[wn + 16 + am][kc * 4 + kb + 1];
            acc[0][0] = wmma4(a0, b0, acc[0][0]);
            acc[0][1] = wmma4(a0, b1, acc[0][1]);
            acc[1][0] = wmma4(a1, b0, acc[1][0]);
            acc[1][1] = wmma4(a1, b1, acc[1][1]);
        }
        __syncthreads();
    }

#pragma unroll
    for (int mt = 0; mt < 2; ++mt) {
        int rowb = m0 + wm + mt * 16 + (lane >> 4) * 8;
#pragma unroll
        for (int nt = 0; nt < 2; ++nt) {
            int col = n0 + wn + nt * 16 + am;
            float bcol = bias[col];
#pragma unroll
            for (int r = 0; r < 8; ++r) {
                size_t idx = (size_t)(rowb + r) * N + col;
                float v = acc[mt][nt][r] + bcol;
                if (resid) v += resid[idx];
                if (do_relu) v = fmaxf(v, 0.f);
                C[idx] = v;
            }
        }
    }
}

// ---------------------------------------------------------------------------
// Fused attention per (graph, head): flash-style online softmax over 256 keys
// with additive bias.  qkv layout: [N, 768] = [q | k | v], head slice = 32.
// Block = 256 threads = 8 waves; wave handles 32 query rows (2 q-tiles).
// ---------------------------------------------------------------------------
__global__ __launch_bounds__(256) void attn_kernel(
    const float* __restrict__ qkv, const float* __restrict__ bias,
    float* __restrict__ out, float scale)
{
    __shared__ __align__(16) float Ks[128][36];
    __shared__ __align__(16) float Vs[128][36];
    __shared__ float Ps[8][16][17];
    int g = blockIdx.x;
    int h = blockIdx.y;
    int tid = threadIdx.x;
    int lane = tid & 31;
    int wave = tid >> 5;
    int q0 = wave * 32;
    int am = lane & 15;
    int kb = (lane >> 4) * 2;

    // Q fragments in registers (A operand, K-dim = head dim 32 -> 8 steps)
    v2f qa[2][8];
#pragma unroll
    for (int qt = 0; qt < 2; ++qt) {
        const float* qp = qkv + (size_t)(g * 256 + q0 + qt * 16 + am) * 768 + h * 32;
#pragma unroll
        for (int kc = 0; kc < 8; ++kc) {
            qa[qt][kc].x = qp[kc * 4 + kb];
            qa[qt][kc].y = qp[kc * 4 + kb + 1];
        }
    }

    float mrow[2][8], lrow[2][8];
    v8f o[2][2];
#pragma unroll
    for (int qt = 0; qt < 2; ++qt) {
#pragma unroll
        for (int r = 0; r < 8; ++r) { mrow[qt][r] = -3e38f; lrow[qt][r] = 0.f; }
        o[qt][0] = vzero8();
        o[qt][1] = vzero8();
    }

    const float* biasGH = bias + ((size_t)(g * 8 + h)) * 256 * 256;

    for (int stage = 0; stage < 2; ++stage) {
        __syncthreads();
        {   // stage 128 K rows + 128 V rows into LDS (async B128 path)
            int lr = tid >> 1;
            int lc = (tid & 1) * 16;
            const float* kp = qkv + (size_t)(g * 256 + stage * 128 + lr) * 768 + 256 + h * 32 + lc;
            const float* vp = kp + 256;
#pragma unroll
            for (int j = 0; j < 4; ++j) {
                g2l_16(kp + j * 4, &Ks[lr][lc + j * 4]);
                g2l_16(vp + j * 4, &Vs[lr][lc + j * 4]);
            }
            wait_async_lds();
        }
        __syncthreads();

        for (int kt = 0; kt < 8; ++kt) {
            int kk0 = kt * 16;
#pragma unroll
            for (int qt = 0; qt < 2; ++qt) {
                // S = Q . K^T  (B[d,kcol] = Ks[kcol][d])
                v8f s = vzero8();
#pragma unroll
                for (int kc = 0; kc < 8; ++kc) {
                    v2f b;
                    b.x = Ks[kk0 + am][kc * 4 + kb];
                    b.y = Ks[kk0 + am][kc * 4 + kb + 1];
                    s = wmma4(qa[qt][kc], b, s);
                }
                int qbase = q0 + qt * 16 + (lane >> 4) * 8;
                int kcol = stage * 128 + kk0 + am;
                float p[8];
#pragma unroll
                for (int r = 0; r < 8; ++r)
                    p[r] = s[r] * scale + biasGH[(size_t)(qbase + r) * 256 + kcol];
                // online softmax update per row
                float corr[8];
#pragma unroll
                for (int r = 0; r < 8; ++r) {
                    float rmax = hmax16(p[r]);
                    float mnew = fmaxf(mrow[qt][r], rmax);
                    corr[r] = __expf(mrow[qt][r] - mnew);
                    p[r] = __expf(p[r] - mnew);
                    lrow[qt][r] = lrow[qt][r] * corr[r] + hsum16(p[r]);
                    mrow[qt][r] = mnew;
                }
#pragma unroll
                for (int dh2 = 0; dh2 < 2; ++dh2)
#pragma unroll
                    for (int r = 0; r < 8; ++r) o[qt][dh2][r] *= corr[r];
                // C-layout P -> LDS -> A-layout (wave-local, DS ops in order)
#pragma unroll
                for (int r = 0; r < 8; ++r) Ps[wave][r + (lane >> 4) * 8][am] = p[r];
                // O += P . V   (B[kcol,d] = Vs[kcol][d])
#pragma unroll
                for (int ks = 0; ks < 4; ++ks) {
                    v2f pa;
                    pa.x = Ps[wave][am][ks * 4 + kb];
                    pa.y = Ps[wave][am][ks * 4 + kb + 1];
#pragma unroll
                    for (int dh2 = 0; dh2 < 2; ++dh2) {
                        v2f vb;
                        int krow = kk0 + ks * 4 + kb;
                        vb.x = Vs[krow][dh2 * 16 + am];
                        vb.y = Vs[krow + 1][dh2 * 16 + am];
                        o[qt][dh2] = wmma4(pa, vb, o[qt][dh2]);
                    }
                }
            }
        }
    }

    // normalize and store: out[(g*256+q), h*32 + d]
#pragma unroll
    for (int qt = 0; qt < 2; ++qt) {
#pragma unroll
        for (int r = 0; r < 8; ++r) {
            float inv = 1.f / lrow[qt][r];
            int qrow = g * 256 + q0 + qt * 16 + r + (lane >> 4) * 8;
            out[(size_t)qrow * 256 + h * 32 + am] = o[qt][0][r] * inv;
            out[(size_t)qrow * 256 + h * 32 + 16 + am] = o[qt][1][r] * inv;
        }
    }
}

// ---------------------------------------------------------------------------
// Sum-pool over nodes of each graph: out[g, c] = sum_q y[g*256+q, c]
// ---------------------------------------------------------------------------
__global__ __launch_bounds__(256) void pool_kernel(
    const float* __restrict__ y, float* __restrict__ out)
{
    int g = blockIdx.x;
    int t = threadIdx.x;
    float acc = 0.f;
    for (int q = 0; q < 256; ++q) acc += y[(size_t)(g * 256 + q) * 256 + t];
    out[g * 256 + t] = acc;
}

// ---------------------------------------------------------------------------
extern "C" void kernel_launch(void* const* d_in, const int* in_sizes, int n_in,
                              void* d_out, int out_size, void* d_ws, size_t ws_size,
                              hipStream_t stream) {
    (void)in_sizes; (void)n_in; (void)out_size; (void)ws_size;
    const int N = 16384, H = 256, G = 64, HEADS = 8, L = 4, F = 512, E = 524288;

    const float* node_embs = (const float*)d_in[0];
    const int* batch_vec   = (const int*)d_in[1];
    const int* ptr         = (const int*)d_in[2];
    const int* edge_index  = (const int*)d_in[3];
    const float* edge_attr = (const float*)d_in[4];
    const float* epw       = (const float*)d_in[5];
    const float* epb       = (const float*)d_in[6];
    const float* Wqkv      = (const float*)d_in[7];
    const float* bqkv      = (const float*)d_in[8];
    const float* Wo        = (const float*)d_in[9];
    const float* bo        = (const float*)d_in[10];
    const float* ln1_g     = (const float*)d_in[11];
    const float* ln1_b     = (const float*)d_in[12];
    const float* ln2_g     = (const float*)d_in[13];
    const float* ln2_b     = (const float*)d_in[14];
    const float* W1        = (const float*)d_in[15];
    const float* b1        = (const float*)d_in[16];
    const float* W2        = (const float*)d_in[17];
    const float* b2        = (const float*)d_in[18];
    const float* norm_g    = (const float*)d_in[19];
    const float* norm_b    = (const float*)d_in[20];
    float* out = (float*)d_out;

    char* ws = (char*)d_ws;
    size_t off = 0;
    float* bias   = (float*)(ws + off); off += (size_t)G * HEADS * 256 * 256 * 4; // 128 MB
    float* x      = (float*)(ws + off); off += (size_t)N * H * 4;                 // 16 MB
    float* y      = (float*)(ws + off); off += (size_t)N * H * 4;                 // 16 MB
    float* qkvb   = (float*)(ws + off); off += (size_t)N * 3 * H * 4;             // 48 MB
    float* attn   = (float*)(ws + off); off += (size_t)N * H * 4;                 // 16 MB
    float* h1     = (float*)(ws + off); off += (size_t)N * F * 4;                 // 32 MB

    (void)hipMemsetAsync(bias, 0, (size_t)G * HEADS * 256 * 256 * 4, stream);
    (void)hipMemcpyAsync(x, node_embs, (size_t)N * H * 4, hipMemcpyDeviceToDevice, stream);

    edge_bias_kernel<<<E / 256, 256, 0, stream>>>(edge_index, edge_attr, epw, epb,
                                                  batch_vec, ptr, bias, E);

    const float scale = 0.17677669529663687f;  // 1/sqrt(32)
    for (int i = 0; i < L; ++i) {
        ln_kernel<<<N, 256, 0, stream>>>(x, ln1_g + i * H, ln1_b + i * H, y);
        gemm_wmma<<<dim3(N / 64, (3 * H) / 128), 256, 0, stream>>>(
            y, Wqkv + (size_t)i * 3 * H * H, bqkv + (size_t)i * 3 * H,
            nullptr, qkvb, N, 3 * H, H, 0);
        attn_kernel<<<dim3(G, HEADS), 256, 0, stream>>>(qkvb, bias, attn, scale);
        gemm_wmma<<<dim3(N / 64, H / 128), 256, 0, stream>>>(
            attn, Wo + (size_t)i * H * H, bo + (size_t)i * H, x, x, N, H, H, 0);
        ln_kernel<<<N, 256, 0, stream>>>(x, ln2_g + i * H, ln2_b + i * H, y);
        gemm_wmma<<<dim3(N / 64, F / 128), 256, 0, stream>>>(
            y, W1 + (size_t)i * F * H, b1 + (size_t)i * F, nullptr, h1, N, F, H, 1);
        gemm_wmma<<<dim3(N / 64, H / 128), 256, 0, stream>>>(
            h1, W2 + (size_t)i * H * F, b2 + (size_t)i * H, x, x, N, H, F, 0);
    }
    ln_kernel<<<N, 256, 0, stream>>>(x, norm_g, norm_b, y);
    pool_kernel<<<G, 256, 0, stream>>>(y, out);
}